// DetectionGenerator_58488864637189
// MI455X (gfx1250) — compile-verified
//
#include <hip/hip_runtime.h>
#include <math.h>

// ---------------- problem constants (from reference) ----------------
#define B_ 8
#define K_ 8192          // 2^13
#define C_ 91
#define CC_ 90
#define CAND_CAP 1024    // per-(b,c) candidate list capacity (sorted)
#define NMS_CAP 256      // candidates entering NMS (top-256 by score)
#define MAX_DET 100
#define SCORE_THRESH 0.05f
#define IOU_THRESH 0.5f
#define BBOX_CLIP 4.135166556742356f  // log(1000/16)

typedef float v2f __attribute__((ext_vector_type(2)));
typedef float v8f __attribute__((ext_vector_type(8)));

// ---------------- workspace layout (bytes) ----------------
#define WS_CNT 0                                   // B*CC ints (candidate counts)
#define WS_CSC 4096                                // B*CC*CAND_CAP floats (cand scores)
#define WS_CID (WS_CSC + B_*CC_*CAND_CAP*4)        // B*CC*CAND_CAP ints   (cand k-index)
#define WS_KSC (WS_CID + B_*CC_*CAND_CAP*4)        // B*CC*MAX_DET floats  (per-class kept scores)
#define WS_KBX (WS_KSC + B_*CC_*MAX_DET*4)         // B*CC*MAX_DET*4 floats (per-class kept boxes)

// ================= kernel 0: zero candidate counters =================
__global__ void det_zero_counts(int* __restrict__ cnt) {
    int i = blockIdx.x * blockDim.x + threadIdx.x;
    if (i < B_ * CC_) cnt[i] = 0;
}

// ================= kernel 1: softmax + threshold push =================
// One wave32 per (b,k) item. Each lane owns classes {lane, lane+32, lane+64}.
__global__ __launch_bounds__(256) void det_softmax_push(
    const float* __restrict__ scores, int* __restrict__ cnt,
    float* __restrict__ csc, int* __restrict__ cid) {
    int wave = threadIdx.x >> 5, lane = threadIdx.x & 31;
    int item = blockIdx.x * 8 + wave;          // item in [0, B*K)
    int b = item >> 13, k = item & (K_ - 1);
    const float* p = scores + (size_t)item * C_;
    __builtin_prefetch(p + 8 * C_, 0, 1);      // gfx1250: global_prefetch_b8

    float x0 = p[lane];
    float x1 = p[lane + 32];
    bool v2 = (lane < C_ - 64);                // lane < 27
    float x2 = v2 ? p[lane + 64] : -1e30f;

    float m = fmaxf(x0, fmaxf(x1, x2));
    #pragma unroll
    for (int o = 16; o; o >>= 1) m = fmaxf(m, __shfl_xor(m, o, 32));
    float e0 = __expf(x0 - m), e1 = __expf(x1 - m), e2 = v2 ? __expf(x2 - m) : 0.f;
    float s = e0 + e1 + e2;
    #pragma unroll
    for (int o = 16; o; o >>= 1) s += __shfl_xor(s, o, 32);
    float inv = 1.f / s;

    // class 0 is background (skipped); class c maps to list (b, c-1)
    float pr0 = e0 * inv;
    if (lane >= 1 && pr0 > SCORE_THRESH) {
        int cc = lane - 1;
        int pos = atomicAdd(&cnt[b * CC_ + cc], 1);
        if (pos < CAND_CAP) { size_t o2 = (size_t)(b * CC_ + cc) * CAND_CAP + pos; csc[o2] = pr0; cid[o2] = k; }
    }
    float pr1 = e1 * inv;
    if (pr1 > SCORE_THRESH) {
        int cc = lane + 31;
        int pos = atomicAdd(&cnt[b * CC_ + cc], 1);
        if (pos < CAND_CAP) { size_t o2 = (size_t)(b * CC_ + cc) * CAND_CAP + pos; csc[o2] = pr1; cid[o2] = k; }
    }
    float pr2 = e2 * inv;
    if (v2 && pr2 > SCORE_THRESH) {
        int cc = lane + 63;
        int pos = atomicAdd(&cnt[b * CC_ + cc], 1);
        if (pos < CAND_CAP) { size_t o2 = (size_t)(b * CC_ + cc) * CAND_CAP + pos; csc[o2] = pr2; cid[o2] = k; }
    }
}

// ===== kernel 2: per-(b,c) sort + decode + WMMA-tiled IoU + greedy NMS =====
__global__ __launch_bounds__(256) void det_nms_class(
    const float* __restrict__ raw_boxes, const float* __restrict__ anchors,
    const float* __restrict__ image_shape,
    const int* __restrict__ cnt, const float* __restrict__ csc, const int* __restrict__ cid,
    float* __restrict__ ksc, float* __restrict__ kbx) {
    __shared__ float s_sc[CAND_CAP];
    __shared__ int   s_id[CAND_CAP];
    __shared__ float s_y1[NMS_CAP], s_x1[NMS_CAP], s_y2[NMS_CAP], s_x2[NMS_CAP], s_ar[NMS_CAP];
    __shared__ unsigned s_mask[NMS_CAP * 8];   // 256x256 suppression bit-matrix
    __shared__ int s_order[MAX_DET];
    __shared__ int s_nout;

    int tid = threadIdx.x;
    int bc = blockIdx.x;                       // 0..B*CC-1
    int b = bc / CC_, c = bc % CC_;
    int n = cnt[bc]; if (n > CAND_CAP) n = CAND_CAP;
    size_t cb = (size_t)bc * CAND_CAP;

    for (int i = tid; i < CAND_CAP; i += 256) {
        if (i < n) { s_sc[i] = csc[cb + i]; s_id[i] = cid[cb + i]; }
        else       { s_sc[i] = -1e30f;      s_id[i] = -1; }
    }
    __syncthreads();

    // ---- bitonic sort descending by score (N = 1024) ----
    for (int sz = 2; sz <= CAND_CAP; sz <<= 1)
        for (int st = sz >> 1; st > 0; st >>= 1) {
            __syncthreads();
            for (int i = tid; i < CAND_CAP; i += 256) {
                int j = i ^ st;
                if (j > i) {
                    bool desc = ((i & sz) == 0);
                    float si = s_sc[i], sj = s_sc[j];
                    if (desc ? (si < sj) : (si > sj)) {
                        s_sc[i] = sj; s_sc[j] = si;
                        int t = s_id[i]; s_id[i] = s_id[j]; s_id[j] = t;
                    }
                }
            }
        }
    __syncthreads();

    int nn = n < NMS_CAP ? n : NMS_CAP;
    float imh = image_shape[b * 2 + 0], imw = image_shape[b * 2 + 1];

    // ---- gather + decode + clip top-nn boxes ----
    for (int i = tid; i < NMS_CAP; i += 256) {
        float y1 = 0.f, x1 = 0.f, y2 = 0.f, x2 = 0.f;
        if (i < nn) {
            int k = s_id[i];
            const float* e = raw_boxes + ((size_t)(b * K_ + k)) * (C_ * 4) + (size_t)(c + 1) * 4;
            const float* a = anchors + ((size_t)(b * K_ + k)) * 4;
            float dy = e[0] * 0.1f, dx = e[1] * 0.1f;
            float dh = fminf(e[2] * 0.2f, BBOX_CLIP), dw = fminf(e[3] * 0.2f, BBOX_CLIP);
            float ah = a[2] - a[0], aw = a[3] - a[1];
            float ayc = a[0] + 0.5f * ah, axc = a[1] + 0.5f * aw;
            float yc = dy * ah + ayc, xc = dx * aw + axc;
            float bh = __expf(dh) * ah, bw = __expf(dw) * aw;
            y1 = fminf(fmaxf(yc - 0.5f * bh, 0.f), imh);
            x1 = fminf(fmaxf(xc - 0.5f * bw, 0.f), imw);
            y2 = fminf(fmaxf(yc + 0.5f * bh, 0.f), imh);
            x2 = fminf(fmaxf(xc + 0.5f * bw, 0.f), imw);
        }
        s_y1[i] = y1; s_x1[i] = x1; s_y2[i] = y2; s_x2[i] = x2;
        s_ar[i] = (y2 - y1) * (x2 - x1);
    }
    for (int i = tid; i < NMS_CAP * 8; i += 256) s_mask[i] = 0u;
    __syncthreads();

    // ---- all-pairs IoU in 16x16 tiles; WMMA computes area[i]+area[j] ----
    int wave = tid >> 5, lane = tid & 31;
    for (int t = wave; t < (NMS_CAP / 16) * (NMS_CAP / 16); t += 8) {
        int ti = t >> 4, tj = t & 15;
        // A (16x4 f32): row i = [area_i, 1, 0, 0]; lanes 0-15 hold K0/K1, 16-31 K2/K3
        // B (4x16 f32): col j = [1, area_j, 0, 0]^T
        v2f Av, Bv; v8f Cacc = {};
        if (lane < 16) {
            Av.x = s_ar[ti * 16 + lane]; Av.y = 1.0f;     // K0 = area_i, K1 = 1
            Bv.x = 1.0f; Bv.y = s_ar[tj * 16 + lane];     // row K0 = 1s, row K1 = area_j
        } else {
            Av.x = 0.f; Av.y = 0.f; Bv.x = 0.f; Bv.y = 0.f;  // K2/K3 rows = 0
        }
        v8f D = __builtin_amdgcn_wmma_f32_16x16x4_f32(
            false, Av, false, Bv, (short)0, Cacc, false, false);

        int j = tj * 16 + (lane & 15);
        int rbase = ti * 16 + ((lane >= 16) ? 8 : 0);
        float jy1 = s_y1[j], jx1 = s_x1[j], jy2 = s_y2[j], jx2 = s_x2[j];
        unsigned wrd = (unsigned)(tj >> 1), sh = (unsigned)((tj & 1) * 16);
        #pragma unroll
        for (int r = 0; r < 8; r++) {
            int i = rbase + r;
            float iy1 = fmaxf(s_y1[i], jy1), ix1 = fmaxf(s_x1[i], jx1);
            float iy2 = fminf(s_y2[i], jy2), ix2 = fminf(s_x2[i], jx2);
            float inter = fmaxf(iy2 - iy1, 0.f) * fmaxf(ix2 - ix1, 0.f);
            float uni = D[r] - inter;                 // area_i + area_j - inter (WMMA result)
            // iou > 0.5  <=>  2*inter > uni   (when uni > 0; when uni <= 0,
            // both boxes are degenerate => inter == 0 => bit false either way).
            bool bit = (inter + inter > uni) && (inter > 0.f) && (j > i);
            unsigned long long bal = __ballot(bit);   // wave32: bits [31:0]
            if (lane == 0) {
                unsigned lo = (unsigned)(bal & 0xFFFFull);         // rows ti*16+r
                unsigned hi = (unsigned)((bal >> 16) & 0xFFFFull); // rows ti*16+r+8
                if (lo) atomicOr(&s_mask[(ti * 16 + r) * 8 + wrd], lo << sh);
                if (hi) atomicOr(&s_mask[(ti * 16 + r + 8) * 8 + wrd], hi << sh);
            }
        }
    }
    __syncthreads();

    // ---- greedy scan over bit-matrix (score order = index order) ----
    if (tid == 0) {
        unsigned remv[8] = {0, 0, 0, 0, 0, 0, 0, 0};
        int out = 0;
        for (int i = 0; i < nn && out < MAX_DET; i++) {
            if (!((remv[i >> 5] >> (i & 31)) & 1u)) {
                s_order[out++] = i;
                #pragma unroll
                for (int q = 0; q < 8; q++) remv[q] |= s_mask[i * 8 + q];
            }
        }
        s_nout = out;
    }
    __syncthreads();

    int nout = s_nout;
    for (int o = tid; o < MAX_DET; o += 256) {
        size_t ob = (size_t)bc * MAX_DET + o;
        if (o < nout) {
            int i = s_order[o];
            ksc[ob] = s_sc[i];
            kbx[ob * 4 + 0] = s_y1[i]; kbx[ob * 4 + 1] = s_x1[i];
            kbx[ob * 4 + 2] = s_y2[i]; kbx[ob * 4 + 3] = s_x2[i];
        } else {
            ksc[ob] = 0.f;
            kbx[ob * 4 + 0] = 0.f; kbx[ob * 4 + 1] = 0.f;
            kbx[ob * 4 + 2] = 0.f; kbx[ob * 4 + 3] = 0.f;
        }
    }
}

// ========== kernel 3: per-batch global top-100 over 9000 entries ==========
__global__ __launch_bounds__(256) void det_final_topk(
    const float* __restrict__ ksc, const float* __restrict__ kbx,
    float* __restrict__ out) {
    __shared__ float s_sc[CC_ * MAX_DET];   // 9000 floats
    __shared__ float r_val[256];
    __shared__ int   r_idx[256];
    __shared__ int   s_cnt;
    int b = blockIdx.x, tid = threadIdx.x;

    for (int i = tid; i < CC_ * MAX_DET; i += 256)
        s_sc[i] = ksc[(size_t)b * CC_ * MAX_DET + i];
    if (tid == 0) s_cnt = 0;
    __syncthreads();

    for (int it = 0; it < MAX_DET; it++) {
        float bv = -1e30f; int bi = 0;
        for (int i = tid; i < CC_ * MAX_DET; i += 256) {
            float v = s_sc[i];
            if (v > bv) { bv = v; bi = i; }
        }
        r_val[tid] = bv; r_idx[tid] = bi;
        __syncthreads();
        for (int st = 128; st > 0; st >>= 1) {
            if (tid < st && r_val[tid + st] > r_val[tid]) {
                r_val[tid] = r_val[tid + st]; r_idx[tid] = r_idx[tid + st];
            }
            __syncthreads();
        }
        if (tid == 0) {
            int idx = r_idx[0]; float v = r_val[0];
            s_sc[idx] = -2e30f;
            size_t src = (size_t)b * CC_ * MAX_DET + idx;
            int slot = b * MAX_DET + it;
            // d_out layout: fin_b[8*100*4] | fin_s[800] | fin_c[800] | num[8]
            out[slot * 4 + 0] = kbx[src * 4 + 0];
            out[slot * 4 + 1] = kbx[src * 4 + 1];
            out[slot * 4 + 2] = kbx[src * 4 + 2];
            out[slot * 4 + 3] = kbx[src * 4 + 3];
            out[3200 + slot] = v;
            out[4000 + slot] = (float)(idx / MAX_DET + 1);
            if (v > -1.0f) s_cnt++;
        }
        __syncthreads();
    }
    if (tid == 0) out[4800 + b] = (float)s_cnt;
}

// ============================ launcher ============================
extern "C" void kernel_launch(void* const* d_in, const int* in_sizes, int n_in,
                              void* d_out, int out_size, void* d_ws, size_t ws_size,
                              hipStream_t stream) {
    const float* raw_boxes   = (const float*)d_in[0];
    const float* raw_scores  = (const float*)d_in[1];
    const float* anchors     = (const float*)d_in[2];
    const float* image_shape = (const float*)d_in[3];
    (void)in_sizes; (void)n_in; (void)out_size; (void)ws_size;

    char* ws = (char*)d_ws;
    int*   cnt = (int*)(ws + WS_CNT);
    float* csc = (float*)(ws + WS_CSC);
    int*   cid = (int*)(ws + WS_CID);
    float* ksc = (float*)(ws + WS_KSC);
    float* kbx = (float*)(ws + WS_KBX);

    det_zero_counts<<<(B_ * CC_ + 255) / 256, 256, 0, stream>>>(cnt);
    det_softmax_push<<<(B_ * K_) / 8, 256, 0, stream>>>(raw_scores, cnt, csc, cid);
    det_nms_class<<<B_ * CC_, 256, 0, stream>>>(raw_boxes, anchors, image_shape,
                                                cnt, csc, cid, ksc, kbx);
    det_final_topk<<<B_, 256, 0, stream>>>(ksc, kbx, (float*)d_out);
}